// GCN_52415780880448
// MI455X (gfx1250) — compile-verified
//
#include <hip/hip_runtime.h>

// ---------------------------------------------------------------------------
// GNN: per-node scalar MLP (1->4->1), mean scatter over 32M edges, scale.
// Edge stream (512MB int64) is moved with CDNA5 async global->LDS loads
// (non-temporal), double-buffered per wave, synced with ASYNCcnt.
// Accumulation uses fire-and-forget agent-scope float/u32 atomics into
// L2-resident 8MB tables.
// ---------------------------------------------------------------------------

#define WAVES_PER_BLOCK 8
#define TILE 128              // edges per tile per wave (must be multiple of 64)

typedef int v4i __attribute__((vector_size(16)));
typedef __attribute__((address_space(1))) v4i gv4i;   // global (prints as __device__)
typedef __attribute__((address_space(3))) v4i lv4i;   // LDS    (prints as __shared__)

__device__ __forceinline__ void async_load_b128(const void* gptr, const void* lptr) {
#if __has_builtin(__builtin_amdgcn_global_load_async_to_lds_b128)
  // (global v4i* src, lds v4i* dst, imm offset, imm cpol); cpol=1 -> TH_LOAD_NT
  __builtin_amdgcn_global_load_async_to_lds_b128(
      (gv4i*)(unsigned long long)gptr,
      (lv4i*)(unsigned long long)lptr,   // low 32 bits of generic == LDS offset
      0, 1);
#else
  unsigned loff = (unsigned)(unsigned long long)lptr;
  asm volatile("global_load_async_to_lds_b128 %0, %1, off th:TH_LOAD_NT"
               :: "v"(loff), "v"(gptr)
               : "memory");
#endif
}

template <int N>
__device__ __forceinline__ void wait_async() {
#if __has_builtin(__builtin_amdgcn_s_wait_asynccnt)
  __builtin_amdgcn_s_wait_asynccnt((unsigned short)N);
#else
  asm volatile("s_wait_asynccnt %0" :: "n"(N));
#endif
  asm volatile("" ::: "memory");   // compiler barrier: LDS reads stay below
}

__device__ __forceinline__ void wait_ds0() {
#if __has_builtin(__builtin_amdgcn_s_wait_dscnt)
  __builtin_amdgcn_s_wait_dscnt(0);
#else
  asm volatile("s_wait_dscnt 0" ::: "memory");
#endif
  asm volatile("" ::: "memory");
}

// ---------------------------------------------------------------------------
// Edge scatter (FIRST in file so the disasm snippet shows the async ops).
// Per-wave double-buffered async pipeline:
//   tile = 128 edges; 4 async b128 ops fill 1KB src idx + 1KB dst idx in LDS.
//   While tile t is processed (gather h[src], 2 atomics/edge), tile t+stride
//   is already in flight (ASYNCcnt <= 4 releases tile t, in-order completion).
// ---------------------------------------------------------------------------
__global__ __launch_bounds__(256) void edge_scatter_kernel(
    const long long* __restrict__ ei, long long E,
    const float* __restrict__ h,
    float* __restrict__ outAcc, unsigned* __restrict__ cnt,
    int nfull)
{
  __shared__ __align__(16) long long sb[WAVES_PER_BLOCK][2][2][TILE]; // [wave][buf][src/dst][edge]
  const int lane = threadIdx.x & 31;
  const int wid  = threadIdx.x >> 5;
  const int gw   = blockIdx.x * WAVES_PER_BLOCK + wid;
  const int nw   = gridDim.x * WAVES_PER_BLOCK;
  const long long* gsrc = ei;
  const long long* gdst = ei + E;

  int t = gw;
  if (t >= nfull) return;   // wave-uniform

  // Prime the pipeline: tile t -> buffer 0.
  {
    const long long* gs = gsrc + (long long)t * TILE;
    const long long* gd = gdst + (long long)t * TILE;
#pragma unroll
    for (int k = 0; k < TILE / 64; ++k) {
      async_load_b128(gs + k * 64 + lane * 2, &sb[wid][0][0][k * 64 + lane * 2]);
      async_load_b128(gd + k * 64 + lane * 2, &sb[wid][0][1][k * 64 + lane * 2]);
    }
  }

  int b = 0;
  for (; t < nfull; t += nw, b ^= 1) {
    const int tn = t + nw;
    if (tn < nfull) {
      wait_ds0();  // prior reads of buffer b^1 retired before overwrite
      const long long* gs = gsrc + (long long)tn * TILE;
      const long long* gd = gdst + (long long)tn * TILE;
#pragma unroll
      for (int k = 0; k < TILE / 64; ++k) {
        async_load_b128(gs + k * 64 + lane * 2, &sb[wid][b ^ 1][0][k * 64 + lane * 2]);
        async_load_b128(gd + k * 64 + lane * 2, &sb[wid][b ^ 1][1][k * 64 + lane * 2]);
      }
      wait_async<(TILE / 64) * 2>();   // <=4 outstanding: tile t's loads done
    } else {
      wait_async<0>();
    }

    // Process tile t from LDS (lane stride 8B -> bank-conflict-free reads,
    // src/dst pairs fuse into ds_load_2addr_b64).
#pragma unroll
    for (int k = 0; k < TILE / 32; ++k) {
      const int i = lane + 32 * k;
      long long s = sb[wid][b][0][i];
      long long d = sb[wid][b][1][i];
      float hv = h[s];                              // 4MB L2-resident gather
      __hip_atomic_fetch_add(&outAcc[d], hv, __ATOMIC_RELAXED, __HIP_MEMORY_SCOPE_AGENT);
      __hip_atomic_fetch_add(&cnt[d], 1u, __ATOMIC_RELAXED, __HIP_MEMORY_SCOPE_AGENT);
    }
  }
}

// Generic edge kernel (grid-strided, plain non-temporal loads). Used for the
// sub-tile tail, and as full fallback when E is odd (b128 alignment of ei+E).
__global__ __launch_bounds__(256) void edge_plain_kernel(
    const long long* __restrict__ ei, long long E, long long start,
    const float* __restrict__ h,
    float* __restrict__ outAcc, unsigned* __restrict__ cnt)
{
  const long long stride = (long long)gridDim.x * blockDim.x;
  for (long long e = start + (long long)blockIdx.x * blockDim.x + threadIdx.x;
       e < E; e += stride) {
    long long s = __builtin_nontemporal_load(ei + e);
    long long d = __builtin_nontemporal_load(ei + E + e);
    float hv = h[s];
    __hip_atomic_fetch_add(&outAcc[d], hv, __ATOMIC_RELAXED, __HIP_MEMORY_SCOPE_AGENT);
    __hip_atomic_fetch_add(&cnt[d], 1u, __ATOMIC_RELAXED, __HIP_MEMORY_SCOPE_AGENT);
  }
}

// ---------------------------------------------------------------------------
// Kernel 1: h[n] = b2 + sum_j W2[j]*relu(W1[j]*x[n]+b1[j]); zero out/cnt.
// ---------------------------------------------------------------------------
__global__ __launch_bounds__(256) void node_mlp_init_kernel(
    const float* __restrict__ x, long long N,
    const float* __restrict__ W1, const float* __restrict__ b1,
    const float* __restrict__ W2, const float* __restrict__ b2,
    float* __restrict__ h, float* __restrict__ outAcc, unsigned* __restrict__ cnt)
{
  long long n = (long long)blockIdx.x * blockDim.x + threadIdx.x;
  if (n >= N) return;
  float xv  = x[n];
  float acc = b2[0];
#pragma unroll
  for (int j = 0; j < 4; ++j)
    acc += W2[j] * fmaxf(W1[j] * xv + b1[j], 0.0f);
  h[n]      = acc;
  outAcc[n] = 0.0f;
  cnt[n]    = 0u;
}

// ---------------------------------------------------------------------------
// Kernel 3: out[n] = (acc / max(cnt,1)) * Wsage
// ---------------------------------------------------------------------------
__global__ __launch_bounds__(256) void finalize_kernel(
    float* __restrict__ out, const unsigned* __restrict__ cnt,
    const float* __restrict__ Wsage, long long N)
{
  long long n = (long long)blockIdx.x * blockDim.x + threadIdx.x;
  if (n >= N) return;
  float c = (float)cnt[n];
  out[n] = out[n] / fmaxf(c, 1.0f) * Wsage[0];
}

// ---------------------------------------------------------------------------
extern "C" void kernel_launch(void* const* d_in, const int* in_sizes, int n_in,
                              void* d_out, int out_size, void* d_ws, size_t ws_size,
                              hipStream_t stream) {
  const float*     x  = (const float*)d_in[0];
  const long long* ei = (const long long*)d_in[1];   // int64 [2, E] flat
  const float*     W1 = (const float*)d_in[2];
  const float*     b1 = (const float*)d_in[3];
  const float*     W2 = (const float*)d_in[4];
  const float*     b2 = (const float*)d_in[5];
  const float*     Ws = (const float*)d_in[6];

  const long long N = in_sizes[0];
  const long long E = (long long)in_sizes[1] / 2;

  float*    out = (float*)d_out;
  float*    h   = (float*)d_ws;                                  // N floats
  unsigned* cnt = (unsigned*)((char*)d_ws + (size_t)N * 4);      // N u32

  const int tpb = 256;
  const int nbN = (int)((N + tpb - 1) / tpb);
  node_mlp_init_kernel<<<nbN, tpb, 0, stream>>>(x, N, W1, b1, W2, b2, h, out, cnt);

  // b128 async loads need 16B alignment of both ei and ei+E -> E must be even.
  const bool can_async = ((E & 1LL) == 0LL);
  long long start = 0;
  if (can_async) {
    const int nfull = (int)(E / TILE);
    if (nfull > 0) {
      int blocks = (nfull + WAVES_PER_BLOCK - 1) / WAVES_PER_BLOCK;
      if (blocks > 2048) blocks = 2048;
      edge_scatter_kernel<<<blocks, 256, 0, stream>>>(ei, E, h, out, cnt, nfull);
    }
    start = (long long)nfull * TILE;
  }
  if (E - start > 0) {
    long long remBlocks = (E - start + tpb - 1) / tpb;
    int blocks = (int)(remBlocks > 2048 ? 2048 : remBlocks);
    edge_plain_kernel<<<blocks, tpb, 0, stream>>>(ei, E, start, h, out, cnt);
  }

  finalize_kernel<<<nbN, tpb, 0, stream>>>(out, cnt, Ws, N);
}